// GraphRecurrentLayerWrapper_43903155699846
// MI455X (gfx1250) — compile-verified
//
#include <hip/hip_runtime.h>

typedef float v2f __attribute__((ext_vector_type(2)));
typedef float v8f __attribute__((ext_vector_type(8)));

#define DDIM 128
#define LDS_PITCH 132   // 128 + 4 pad: A-tile rows land on distinct LDS banks

// ---------------------------------------------------------------------------
// GEMM:  Tout[N,128] = A0[N,128] @ B[0:128,:]  (+ A1[N,128] @ B[128:256,:])
// One block = 16 output rows, 8 waves each owning a 16-col slab.
// B slab is preloaded into 64 VGPRs per phase (weights are shared by every
// block), so the K-loop is pure ds_load_b64 + v_wmma_f32_16x16x4_f32.
// ---------------------------------------------------------------------------
__global__ __launch_bounds__(256)
void gemm_wmma_f32(const float* __restrict__ A0,
                   const float* __restrict__ A1,   // nullptr => K=128
                   const float* __restrict__ B,    // row-major [K,128]
                   float* __restrict__ Tout,
                   int N)
{
    __shared__ float lds[16 * LDS_PITCH];

    const int tid     = threadIdx.x;
    const int lane    = tid & 31;
    const int wave    = tid >> 5;
    const int rowBase = blockIdx.x * 16;
    const int colBase = wave * 16;
    const int mrow    = lane & 15;          // M (for A) / N (for B) within tile
    const int khalf   = (lane >> 4) * 2;    // lanes 16-31 carry K+2,K+3
    const int col     = colBase + mrow;

    v8f acc = {};

    const float* Aptr = A0;
    const float* Bptr = B;
    const int phases  = (A1 != nullptr) ? 2 : 1;

    for (int p = 0; p < phases; ++p) {
        // ---- preload this wave's B slab into registers (64 VGPRs) ----
        float breg[64];
        #pragma unroll
        for (int i = 0; i < 32; ++i) {      // i = k/4
            const float* bp = Bptr + (size_t)(4 * i + khalf) * DDIM + col;
            breg[2 * i]     = bp[0];
            breg[2 * i + 1] = bp[DDIM];
        }

        // ---- cooperative 16x128 A-tile load (zero-pad rows beyond N) ----
        __syncthreads();   // protect LDS against reuse across phases
        for (int i = tid; i < 16 * DDIM; i += 256) {
            int r  = i >> 7;
            int c  = i & 127;
            int gr = rowBase + r;
            lds[r * LDS_PITCH + c] = (gr < N) ? Aptr[(size_t)gr * DDIM + c] : 0.0f;
        }
        __syncthreads();

        // ---- K loop: pure LDS + WMMA ----
        #pragma unroll
        for (int i = 0; i < 32; ++i) {
            const float* ap = &lds[mrow * LDS_PITCH + 4 * i + khalf];
            v2f a; a.x = ap[0];          a.y = ap[1];
            v2f b; b.x = breg[2 * i];    b.y = breg[2 * i + 1];
            acc = __builtin_amdgcn_wmma_f32_16x16x4_f32(
                false, a, false, b, (short)0, acc, false, false);
        }

        Aptr = A1;
        Bptr = B + (size_t)DDIM * DDIM;   // rows 128..255 of concat weight
    }

    // ---- store: VGPR r -> M=r (lanes 0-15), M=r+8 (lanes 16-31) ----
    const int rofs = (lane >> 4) * 8;
    if (rowBase + 16 <= N) {
        // full tile: unguarded coalesced stores
        #pragma unroll
        for (int r = 0; r < 8; ++r)
            Tout[(size_t)(rowBase + r + rofs) * DDIM + col] = acc[r];
    } else {
        #pragma unroll
        for (int r = 0; r < 8; ++r) {
            int row = rowBase + r + rofs;
            if (row < N) Tout[(size_t)row * DDIM + col] = acc[r];
        }
    }
}

// ---------------------------------------------------------------------------
// Scatter-add over edges: agg[dst[e],:] += T[src[e],:]
// 32 lanes per edge, float4 per lane -> one coalesced 512B row read,
// fp32 hardware atomics into the (L2-resident) aggregation buffer.
// ---------------------------------------------------------------------------
__global__ __launch_bounds__(256)
void scatter_add_edges(const float* __restrict__ T,
                       const int* __restrict__ src,
                       const int* __restrict__ dst,
                       float* __restrict__ agg,
                       int E)
{
    long long gid = (long long)blockIdx.x * blockDim.x + threadIdx.x;
    int e = (int)(gid >> 5);
    if (e >= E) return;
    int c = ((int)gid & 31) * 4;

    int s = src[e];
    int d = dst[e];
    const float4 v = *(const float4*)(T + (size_t)s * DDIM + c);
    float* o = agg + (size_t)d * DDIM + c;
    unsafeAtomicAdd(o + 0, v.x);
    unsafeAtomicAdd(o + 1, v.y);
    unsafeAtomicAdd(o + 2, v.z);
    unsafeAtomicAdd(o + 3, v.w);
}

// ---------------------------------------------------------------------------
// Elementwise epilogues (float4-vectorized: pure-bandwidth kernels)
// ---------------------------------------------------------------------------
__device__ __forceinline__ float sigmoidf_(float x) {
    return 1.0f / (1.0f + __expf(-x));
}

// X = relu(agg + b)
__global__ __launch_bounds__(256)
void relu_bias_kernel(const float4* __restrict__ agg, const float4* __restrict__ b,
                      float4* __restrict__ out, int total4)
{
    int i = blockIdx.x * blockDim.x + threadIdx.x;
    if (i >= total4) return;
    float4 v = agg[i];
    float4 bb = b[i & 31];
    v.x = fmaxf(v.x + bb.x, 0.0f);
    v.y = fmaxf(v.y + bb.y, 0.0f);
    v.z = fmaxf(v.z + bb.z, 0.0f);
    v.w = fmaxf(v.w + bb.w, 0.0f);
    out[i] = v;
}

// Z = sigmoid(agg + b)
__global__ __launch_bounds__(256)
void sigmoid_bias_kernel(const float4* __restrict__ agg, const float4* __restrict__ b,
                         float4* __restrict__ out, int total4)
{
    int i = blockIdx.x * blockDim.x + threadIdx.x;
    if (i >= total4) return;
    float4 v = agg[i];
    float4 bb = b[i & 31];
    v.x = sigmoidf_(v.x + bb.x);
    v.y = sigmoidf_(v.y + bb.y);
    v.z = sigmoidf_(v.z + bb.z);
    v.w = sigmoidf_(v.w + bb.w);
    out[i] = v;
}

// RH = sigmoid(agg + b) * H_prev   (fuses R and R*H_prev)
__global__ __launch_bounds__(256)
void sigmoid_mul_kernel(const float4* __restrict__ agg, const float4* __restrict__ b,
                        const float4* __restrict__ H, float4* __restrict__ out, int total4)
{
    int i = blockIdx.x * blockDim.x + threadIdx.x;
    if (i >= total4) return;
    float4 v = agg[i];
    float4 bb = b[i & 31];
    float4 h = H[i];
    v.x = sigmoidf_(v.x + bb.x) * h.x;
    v.y = sigmoidf_(v.y + bb.y) * h.y;
    v.z = sigmoidf_(v.z + bb.z) * h.z;
    v.w = sigmoidf_(v.w + bb.w) * h.w;
    out[i] = v;
}

// H_out = Z*H_prev + (1-Z)*tanh(agg + b)
__global__ __launch_bounds__(256)
void final_gate_kernel(const float4* __restrict__ agg, const float4* __restrict__ b,
                       const float4* __restrict__ Z, const float4* __restrict__ H,
                       float4* __restrict__ out, int total4)
{
    int i = blockIdx.x * blockDim.x + threadIdx.x;
    if (i >= total4) return;
    float4 v = agg[i];
    float4 bb = b[i & 31];
    float4 z = Z[i];
    float4 h = H[i];
    v.x = z.x * h.x + (1.0f - z.x) * tanhf(v.x + bb.x);
    v.y = z.y * h.y + (1.0f - z.y) * tanhf(v.y + bb.y);
    v.z = z.z * h.z + (1.0f - z.z) * tanhf(v.z + bb.z);
    v.w = z.w * h.w + (1.0f - z.w) * tanhf(v.w + bb.w);
    out[i] = v;
}

// ---------------------------------------------------------------------------
// Orchestration
// ---------------------------------------------------------------------------
extern "C" void kernel_launch(void* const* d_in, const int* in_sizes, int n_in,
                              void* d_out, int out_size, void* d_ws, size_t ws_size,
                              hipStream_t stream)
{
    const float* nodef = (const float*)d_in[0];
    const float* Hprev = (const float*)d_in[1];
    const int*   eidx  = (const int*)  d_in[2];
    const float* Wm    = (const float*)d_in[3];
    const float* bm    = (const float*)d_in[4];
    const float* Wz    = (const float*)d_in[5];
    const float* bz    = (const float*)d_in[6];
    const float* Wr    = (const float*)d_in[7];
    const float* br    = (const float*)d_in[8];
    const float* Wh    = (const float*)d_in[9];
    const float* bh    = (const float*)d_in[10];

    const int N = in_sizes[0] / DDIM;
    const int E = in_sizes[2] / 2;
    const int* src = eidx;       // edge_index[0,:]
    const int* dst = eidx + E;   // edge_index[1,:]

    const size_t nd = (size_t)N * DDIM;
    float* T   = (float*)d_ws;
    float* AGG = T   + nd;
    float* X   = AGG + nd;
    float* Z   = X   + nd;
    float* RH  = Z   + nd;
    float* Hout = (float*)d_out;

    const int total4    = (int)(nd / 4);
    const int gemmGrid  = (N + 15) / 16;
    const int ewGrid    = (total4 + 255) / 256;
    const int scGrid    = (int)(((long long)E * 32 + 255) / 256);
    const size_t aggBytes = nd * sizeof(float);

    #define F4(p) ((float4*)(p))
    #define CF4(p) ((const float4*)(p))

    // ---- main conv: X = relu(scatter(nodef @ Wm) + bm) ----
    gemm_wmma_f32<<<gemmGrid, 256, 0, stream>>>(nodef, nullptr, Wm, T, N);
    hipMemsetAsync(AGG, 0, aggBytes, stream);
    scatter_add_edges<<<scGrid, 256, 0, stream>>>(T, src, dst, AGG, E);
    relu_bias_kernel<<<ewGrid, 256, 0, stream>>>(CF4(AGG), CF4(bm), F4(X), total4);

    // ---- Z = sigmoid(scatter([X|H] @ Wz) + bz) ----
    gemm_wmma_f32<<<gemmGrid, 256, 0, stream>>>(X, Hprev, Wz, T, N);
    hipMemsetAsync(AGG, 0, aggBytes, stream);
    scatter_add_edges<<<scGrid, 256, 0, stream>>>(T, src, dst, AGG, E);
    sigmoid_bias_kernel<<<ewGrid, 256, 0, stream>>>(CF4(AGG), CF4(bz), F4(Z), total4);

    // ---- RH = sigmoid(scatter([X|H] @ Wr) + br) * H ----
    gemm_wmma_f32<<<gemmGrid, 256, 0, stream>>>(X, Hprev, Wr, T, N);
    hipMemsetAsync(AGG, 0, aggBytes, stream);
    scatter_add_edges<<<scGrid, 256, 0, stream>>>(T, src, dst, AGG, E);
    sigmoid_mul_kernel<<<ewGrid, 256, 0, stream>>>(CF4(AGG), CF4(br), CF4(Hprev), F4(RH), total4);

    // ---- H_out = Z*H + (1-Z)*tanh(scatter([X|RH] @ Wh) + bh) ----
    gemm_wmma_f32<<<gemmGrid, 256, 0, stream>>>(X, RH, Wh, T, N);
    hipMemsetAsync(AGG, 0, aggBytes, stream);
    scatter_add_edges<<<scGrid, 256, 0, stream>>>(T, src, dst, AGG, E);
    final_gate_kernel<<<ewGrid, 256, 0, stream>>>(CF4(AGG), CF4(bh), CF4(Z), CF4(Hprev), F4(Hout), total4);
}